// EdgeConditionedLayerV2_41223096107203
// MI455X (gfx1250) — compile-verified
//
#include <hip/hip_runtime.h>
#include <hip/hip_bf16.h>

typedef float v2f __attribute__((ext_vector_type(2)));
typedef float v8f __attribute__((ext_vector_type(8)));

#define N_NODES 50000
#define N_EDGES 800000
#define DIM 128
#define EFEAT 16
#define HID 32

// ---------------------------------------------------------------------------
// zero fill (grid-stride)
__global__ void zero_kernel(float* __restrict__ p, long n) {
    long i = (long)blockIdx.x * blockDim.x + threadIdx.x;
    long stride = (long)gridDim.x * blockDim.x;
    for (; i < n; i += stride) p[i] = 0.0f;
}

// ---------------------------------------------------------------------------
// edge MLP: ew[e] = sigmoid(relu(attr @ We1 + be1) @ We2 + be2)
__global__ void edge_mlp_kernel(const float* __restrict__ attr,
                                const float* __restrict__ We1,
                                const float* __restrict__ be1,
                                const float* __restrict__ We2,
                                const float* __restrict__ be2,
                                float* __restrict__ ew, int E) {
    __shared__ float sW1[EFEAT * HID];
    __shared__ float sb1[HID];
    __shared__ float sW2[HID];
    __shared__ float sb2;
    for (int i = threadIdx.x; i < EFEAT * HID; i += blockDim.x) sW1[i] = We1[i];
    if (threadIdx.x < HID) {
        sb1[threadIdx.x] = be1[threadIdx.x];
        sW2[threadIdx.x] = We2[threadIdx.x];
    }
    if (threadIdx.x == 0) sb2 = be2[0];
    __syncthreads();

    int e = blockIdx.x * blockDim.x + threadIdx.x;
    if (e >= E) return;

    float a[EFEAT];
    const float4* ap = (const float4*)(attr + (long)e * EFEAT);
#pragma unroll
    for (int q = 0; q < 4; q++) {
        float4 v = ap[q];
        a[q * 4 + 0] = v.x; a[q * 4 + 1] = v.y; a[q * 4 + 2] = v.z; a[q * 4 + 3] = v.w;
    }
    float acc = sb2;
#pragma unroll
    for (int h = 0; h < HID; h++) {
        float s = sb1[h];
#pragma unroll
        for (int k = 0; k < EFEAT; k++) s = fmaf(a[k], sW1[k * HID + h], s);
        s = fmaxf(s, 0.0f);
        acc = fmaf(s, sW2[h], acc);
    }
    ew[e] = 1.0f / (1.0f + __expf(-acc));
}

// ---------------------------------------------------------------------------
// symmetric degree accumulation: deg[src]+=ew, deg[dst]+=ew
__global__ void degree_kernel(const int* __restrict__ src, const int* __restrict__ dst,
                              const float* __restrict__ ew, float* __restrict__ deg, int E) {
    int e = blockIdx.x * blockDim.x + threadIdx.x;
    if (e >= E) return;
    float w = ew[e];
    unsafeAtomicAdd(&deg[src[e]], w);
    unsafeAtomicAdd(&deg[dst[e]], w);
}

// dinv[i] = (deg[i] + eps)^-0.5
__global__ void dinv_kernel(const float* __restrict__ deg, float* __restrict__ dinv, int n) {
    int i = blockIdx.x * blockDim.x + threadIdx.x;
    if (i < n) dinv[i] = __frsqrt_rn(deg[i] + 1e-12f);
}

// vals[e] = ew[e] * dinv[src] * dinv[dst]   (symmetric: same both directions)
__global__ void edge_norm_kernel(const int* __restrict__ src, const int* __restrict__ dst,
                                 const float* __restrict__ ew, const float* __restrict__ dinv,
                                 float* __restrict__ vals, int E) {
    int e = blockIdx.x * blockDim.x + threadIdx.x;
    if (e >= E) return;
    vals[e] = ew[e] * dinv[src[e]] * dinv[dst[e]];
}

// ---------------------------------------------------------------------------
// dual WMMA f32 GEMM: HW1 = H@W1, HW2 = H@W2. grid.x = M/16 tiles, 256 thr (8 waves),
// wave w owns columns [16w, 16w+16). K = 128 stepped by 4 (V_WMMA_F32_16X16X4_F32).
__global__ void gemm_dual_kernel(const float* __restrict__ H,
                                 const float* __restrict__ W1,
                                 const float* __restrict__ W2,
                                 float* __restrict__ HW1, float* __restrict__ HW2) {
    const int wave = threadIdx.x >> 5;
    const int lane = threadIdx.x & 31;
    const int half = lane >> 4;   // 0: K pair {0,1} / rows M..  1: K pair {2,3} / rows M+8..
    const int l = lane & 15;
    const int m0 = blockIdx.x * 16;
    const int n0 = wave * 16;
    const long arow = m0 + l;
    const int bcol = n0 + l;

    v8f c1 = {};
    v8f c2 = {};
#pragma unroll
    for (int k0 = 0; k0 < DIM; k0 += 4) {
        const int ka = k0 + half * 2;
        v2f a;
        a.x = H[arow * DIM + ka];
        a.y = H[arow * DIM + ka + 1];
        v2f b1, b2;
        b1.x = W1[(long)ka * DIM + bcol];
        b1.y = W1[(long)(ka + 1) * DIM + bcol];
        b2.x = W2[(long)ka * DIM + bcol];
        b2.y = W2[(long)(ka + 1) * DIM + bcol];
        c1 = __builtin_amdgcn_wmma_f32_16x16x4_f32(false, a, false, b1, (short)0, c1, false, false);
        c2 = __builtin_amdgcn_wmma_f32_16x16x4_f32(false, a, false, b2, (short)0, c2, false, false);
    }
#pragma unroll
    for (int v = 0; v < 8; v++) {
        const long row = m0 + v + half * 8;
        HW1[row * DIM + bcol] = c1[v];
        HW2[row * DIM + bcol] = c2[v];
    }
}

// ---------------------------------------------------------------------------
// SpMM scatter: one wave per edge; both symmetric directions.
// out[s,:] += v*x[d,:]; out[d,:] += v*x[s,:]
__global__ void spmm_kernel(const int* __restrict__ src, const int* __restrict__ dst,
                            const float* __restrict__ vals, const float* __restrict__ x,
                            float* __restrict__ out, int E) {
    const int gwave = (blockIdx.x * blockDim.x + threadIdx.x) >> 5;
    const int lane = threadIdx.x & 31;
    if (gwave >= E) return;
    const int s = src[gwave];
    const int d = dst[gwave];
    const float v = vals[gwave];
    const float4 xd = ((const float4*)(x + (long)d * DIM))[lane];
    const float4 xs = ((const float4*)(x + (long)s * DIM))[lane];
    float* os = out + (long)s * DIM + lane * 4;
    float* od = out + (long)d * DIM + lane * 4;
    unsafeAtomicAdd(os + 0, v * xd.x);
    unsafeAtomicAdd(os + 1, v * xd.y);
    unsafeAtomicAdd(os + 2, v * xd.z);
    unsafeAtomicAdd(os + 3, v * xd.w);
    unsafeAtomicAdd(od + 0, v * xs.x);
    unsafeAtomicAdd(od + 1, v * xs.y);
    unsafeAtomicAdd(od + 2, v * xs.z);
    unsafeAtomicAdd(od + 3, v * xs.w);
}

// ---------------------------------------------------------------------------
// fused epilogue GEMM: out = relu([gc*h1+(1-gc)*h2 | H] @ W_up + b_up)
// K = 256: k<128 reads gated aggregate, k>=128 reads H. WMMA f32 16x16x4.
__global__ void final_gemm_kernel(const float* __restrict__ h1, const float* __restrict__ h2,
                                  const float* __restrict__ Hin,
                                  const float* __restrict__ Wup, const float* __restrict__ bup,
                                  const float* __restrict__ gptr, float* __restrict__ out) {
    const int wave = threadIdx.x >> 5;
    const int lane = threadIdx.x & 31;
    const int half = lane >> 4;
    const int l = lane & 15;
    const int m0 = blockIdx.x * 16;
    const int n0 = wave * 16;
    const long arow = m0 + l;
    const int bcol = n0 + l;

    float gc = gptr[0];
    gc = fminf(fmaxf(gc, 0.0f), 1.0f);
    const float gi = 1.0f - gc;

    v8f c = {};
    // k in [0,128): A = gc*h1 + (1-gc)*h2
#pragma unroll
    for (int k0 = 0; k0 < DIM; k0 += 4) {
        const int ka = k0 + half * 2;
        v2f a, b;
        a.x = gc * h1[arow * DIM + ka]     + gi * h2[arow * DIM + ka];
        a.y = gc * h1[arow * DIM + ka + 1] + gi * h2[arow * DIM + ka + 1];
        b.x = Wup[(long)ka * DIM + bcol];
        b.y = Wup[(long)(ka + 1) * DIM + bcol];
        c = __builtin_amdgcn_wmma_f32_16x16x4_f32(false, a, false, b, (short)0, c, false, false);
    }
    // k in [128,256): A = H
#pragma unroll
    for (int k0 = DIM; k0 < 2 * DIM; k0 += 4) {
        const int ka = k0 + half * 2;
        v2f a, b;
        a.x = Hin[arow * DIM + (ka - DIM)];
        a.y = Hin[arow * DIM + (ka - DIM + 1)];
        b.x = Wup[(long)ka * DIM + bcol];
        b.y = Wup[(long)(ka + 1) * DIM + bcol];
        c = __builtin_amdgcn_wmma_f32_16x16x4_f32(false, a, false, b, (short)0, c, false, false);
    }
    const float bb = bup[bcol];
#pragma unroll
    for (int v = 0; v < 8; v++) {
        const long row = m0 + v + half * 8;
        out[row * DIM + bcol] = fmaxf(c[v] + bb, 0.0f);
    }
}

// ---------------------------------------------------------------------------
extern "C" void kernel_launch(void* const* d_in, const int* in_sizes, int n_in,
                              void* d_out, int out_size, void* d_ws, size_t ws_size,
                              hipStream_t stream) {
    const float* H    = (const float*)d_in[0];
    const int*   eidx = (const int*)d_in[1];
    const float* attr = (const float*)d_in[2];
    const float* W1   = (const float*)d_in[3];
    const float* W2   = (const float*)d_in[4];
    const float* We1  = (const float*)d_in[5];
    const float* be1  = (const float*)d_in[6];
    const float* We2  = (const float*)d_in[7];
    const float* be2  = (const float*)d_in[8];
    const float* g    = (const float*)d_in[9];
    const float* Wup  = (const float*)d_in[10];
    const float* bup  = (const float*)d_in[11];
    float* out = (float*)d_out;

    const int N = N_NODES;
    const int E = N_EDGES;
    const int* src = eidx;       // edge_index[0,:]
    const int* dst = eidx + E;   // edge_index[1,:]

    // workspace carve-up (floats)
    float* ws   = (float*)d_ws;
    float* ew   = ws;                         // E
    float* deg  = ew + E;                     // N
    float* dinv = deg + N;                    // N
    float* vals = dinv + N;                   // E
    float* HW1  = vals + E;                   // N*DIM
    float* HW2  = HW1 + (long)N * DIM;        // N*DIM
    float* h1   = HW2 + (long)N * DIM;        // N*DIM
    float* tmp  = h1 + (long)N * DIM;         // N*DIM
    float* h2   = tmp + (long)N * DIM;        // N*DIM

    // 1) zero accumulators: deg, and h1|tmp|h2 (contiguous)
    zero_kernel<<<1024, 256, 0, stream>>>(deg, (long)N);
    zero_kernel<<<4096, 256, 0, stream>>>(h1, 3L * N * DIM);

    // 2) edge MLP -> sigmoid weights
    edge_mlp_kernel<<<(E + 255) / 256, 256, 0, stream>>>(attr, We1, be1, We2, be2, ew, E);

    // 3) symmetric degree, 4) dinv, 5) normalized edge values
    degree_kernel<<<(E + 255) / 256, 256, 0, stream>>>(src, dst, ew, deg, E);
    dinv_kernel<<<(N + 255) / 256, 256, 0, stream>>>(deg, dinv, N);
    edge_norm_kernel<<<(E + 255) / 256, 256, 0, stream>>>(src, dst, ew, dinv, vals, E);

    // 6) dense projections HW1 = H@W1, HW2 = H@W2 (WMMA f32)
    gemm_dual_kernel<<<N / 16, 256, 0, stream>>>(H, W1, W2, HW1, HW2);

    // 7) SpMMs: h1 = A@HW1 ; tmp = A@HW2 ; h2 = A@tmp
    const int spmm_blocks = (E + 7) / 8;   // 8 waves (edges) per 256-thread block
    spmm_kernel<<<spmm_blocks, 256, 0, stream>>>(src, dst, vals, HW1, h1, E);
    spmm_kernel<<<spmm_blocks, 256, 0, stream>>>(src, dst, vals, HW2, tmp, E);
    spmm_kernel<<<spmm_blocks, 256, 0, stream>>>(src, dst, vals, tmp, h2, E);

    // 8) fused gate + concat GEMM + bias + relu -> out
    final_gemm_kernel<<<N / 16, 256, 0, stream>>>(h1, h2, H, Wup, bup, g, out);
}